// GCN_77240691851645
// MI455X (gfx1250) — compile-verified
//
#include <hip/hip_runtime.h>
#include <hip/hip_bf16.h>

typedef __attribute__((ext_vector_type(2))) float v2f;
typedef __attribute__((ext_vector_type(8))) float v8f;

// ---------------------------------------------------------------------------
// Zero (float4 granularity)
// ---------------------------------------------------------------------------
__global__ void gcn_zero4(float4* __restrict__ p, int n4) {
    int i = blockIdx.x * blockDim.x + threadIdx.x;
    if (i < n4) p[i] = make_float4(0.f, 0.f, 0.f, 0.f);
}

// ---------------------------------------------------------------------------
// Degree accumulation: out_deg[src[e]] += 1, in_deg[dst[e]] += 1
// ---------------------------------------------------------------------------
__global__ void gcn_degrees(const int* __restrict__ src, const int* __restrict__ dst,
                            float* __restrict__ outdeg, float* __restrict__ indeg, int E) {
    int e = blockIdx.x * blockDim.x + threadIdx.x;
    if (e < E) {
        atomicAdd(outdeg + src[e], 1.0f);
        atomicAdd(indeg + dst[e], 1.0f);
    }
}

// ns = deg>0 ? rsqrt(deg) : 0   (deg is an integer-valued float >= 1 when >0)
__global__ void gcn_norms(const float* __restrict__ outdeg, const float* __restrict__ indeg,
                          float* __restrict__ ns, float* __restrict__ nd, int N) {
    int i = blockIdx.x * blockDim.x + threadIdx.x;
    if (i < N) {
        float od = outdeg[i];
        float id = indeg[i];
        ns[i] = od > 0.f ? rsqrtf(od) : 0.f;
        nd[i] = id > 0.f ? rsqrtf(id) : 0.f;
    }
}

// ---------------------------------------------------------------------------
// Dense transform: H[N,OUTF] = (X[N,K] * ns[:,None]) @ W[K,OUTF]
// WMMA f32 16x16x4. One 16x16 C tile per wave; block = 2 row-tiles x CT col-tiles.
// A/W staged in LDS with padded strides to dodge bank conflicts.
// EXEC is never restricted before the WMMA (ISA requirement).
// ---------------------------------------------------------------------------
template <int K, int OUTF>
__global__ void gcn_gemm_wmma(const float* __restrict__ X, const float* __restrict__ ns,
                              const float* __restrict__ W, float* __restrict__ H, int N) {
    constexpr int CT  = (OUTF + 15) / 16;  // col tiles
    constexpr int LDA = K + 4;             // sA row stride (floats), breaks 64-bank alias
    constexpr int LDW = 72;                // sW row stride (floats), 2*72 mod 64 = 16
    __shared__ float sA[32 * LDA];
    __shared__ float sW[K * LDW];

    const int tid      = threadIdx.x;
    const int nthreads = 32 * 2 * CT;
    const int rowBase  = blockIdx.x * 32;

    // Stage A = X * ns  (32 rows x K), zero-padded past N
    for (int idx = tid; idx < 32 * K; idx += nthreads) {
        int m = idx / K, k = idx % K;
        int row = rowBase + m;
        float v = 0.f;
        if (row < N) v = X[(size_t)row * K + k] * ns[row];
        sA[m * LDA + k] = v;
    }
    // Stage W (K x OUTF), padded to 64 cols with zeros
    for (int idx = tid; idx < K * 64; idx += nthreads) {
        int k = idx / 64, j = idx % 64;
        sW[k * LDW + j] = (j < OUTF) ? W[k * OUTF + j] : 0.f;
    }
    __syncthreads();

    const int lane  = tid & 31;
    const int wave  = tid >> 5;
    const int ct    = wave % CT;        // col tile
    const int rt    = wave / CT;        // row tile (0..1)
    const int mBase = rt * 16;
    const int half  = lane >> 4;        // 0: lanes 0-15, 1: lanes 16-31
    const int l16   = lane & 15;
    const int col   = ct * 16 + l16;

    v8f acc = {0.f, 0.f, 0.f, 0.f, 0.f, 0.f, 0.f, 0.f};

    // A frag (16x4): lane l16 = row, half selects K pair; a[v] = A[m][k0+2*half+v]
    // B frag (4x16): lane l16 = col;                       b[v] = B[k0+2*half+v][col]
    for (int k0 = 0; k0 < K; k0 += 4) {
        const int ka = k0 + 2 * half;
        v2f a, b;
        a.x = sA[(mBase + l16) * LDA + ka];
        a.y = sA[(mBase + l16) * LDA + ka + 1];
        b.x = sW[(ka + 0) * LDW + col];
        b.y = sW[(ka + 1) * LDW + col];
        acc = __builtin_amdgcn_wmma_f32_16x16x4_f32(
            /*neg_a=*/false, a, /*neg_b=*/false, b,
            /*c_mod=*/(short)0, acc, /*reuse_a=*/false, /*reuse_b=*/false);
    }

    // C layout: VGPR r, lanes 0-15 -> M=r, lanes 16-31 -> M=r+8
#pragma unroll
    for (int r = 0; r < 8; ++r) {
        int row = rowBase + mBase + r + half * 8;
        if (row < N && col < OUTF) H[(size_t)row * OUTF + col] = acc[r];
    }
}

// ---------------------------------------------------------------------------
// SpMM scatter: agg[dst[e]] += H[src[e]]   (float4 gather, 4x f32 atomics)
// H/agg tables (<=25.6 MB) are L2-resident on MI455X (192 MB L2).
// ---------------------------------------------------------------------------
template <int OUTF>
__global__ void gcn_scatter(const float* __restrict__ H, const int* __restrict__ src,
                            const int* __restrict__ dst, float* __restrict__ agg, int E) {
    constexpr int VPE = OUTF / 4;  // float4 chunks per edge
    int idx = blockIdx.x * blockDim.x + threadIdx.x;
    int total = E * VPE;
    if (idx >= total) return;
    int e = idx / VPE;
    int c = (idx % VPE) * 4;
    int s = __ldg(src + e);
    int d = __ldg(dst + e);
    const float4 v = *reinterpret_cast<const float4*>(H + (size_t)s * OUTF + c);
    float* p = agg + (size_t)d * OUTF + c;
    atomicAdd(p + 0, v.x);
    atomicAdd(p + 1, v.y);
    atomicAdd(p + 2, v.z);
    atomicAdd(p + 3, v.w);
}

// ---------------------------------------------------------------------------
// Epilogue: out = relu?(agg * nd[:,None] + b)
// ---------------------------------------------------------------------------
template <int OUTF, bool RELU>
__global__ void gcn_finalize(const float* __restrict__ agg, const float* __restrict__ nd,
                             const float* __restrict__ bias, float* __restrict__ out, int N) {
    constexpr int VPN = OUTF / 4;
    int idx = blockIdx.x * blockDim.x + threadIdx.x;
    int total = N * VPN;
    if (idx >= total) return;
    int row = idx / VPN;
    int c = (idx % VPN) * 4;
    float4 a = *reinterpret_cast<const float4*>(agg + (size_t)row * OUTF + c);
    float4 b = *reinterpret_cast<const float4*>(bias + c);
    float s = nd[row];
    float4 o = make_float4(fmaf(a.x, s, b.x), fmaf(a.y, s, b.y),
                           fmaf(a.z, s, b.z), fmaf(a.w, s, b.w));
    if (RELU) {
        o.x = fmaxf(o.x, 0.f); o.y = fmaxf(o.y, 0.f);
        o.z = fmaxf(o.z, 0.f); o.w = fmaxf(o.w, 0.f);
    }
    *reinterpret_cast<float4*>(out + (size_t)row * OUTF + c) = o;
}

// ---------------------------------------------------------------------------
static inline int cdiv(long long a, long long b) { return (int)((a + b - 1) / b); }

extern "C" void kernel_launch(void* const* d_in, const int* in_sizes, int n_in,
                              void* d_out, int out_size, void* d_ws, size_t ws_size,
                              hipStream_t stream) {
    const float* feat = (const float*)d_in[0];
    const int*   src  = (const int*)d_in[1];
    const int*   dst  = (const int*)d_in[2];
    const float* W0 = (const float*)d_in[3];  const float* b0 = (const float*)d_in[4];
    const float* W1 = (const float*)d_in[5];  const float* b1 = (const float*)d_in[6];
    const float* W2 = (const float*)d_in[7];  const float* b2 = (const float*)d_in[8];
    const float* W3 = (const float*)d_in[9];  const float* b3 = (const float*)d_in[10];

    const int N = in_sizes[0] / 128;  // 100000
    const int E = in_sizes[1];        // 1600000

    float* ws     = (float*)d_ws;
    float* outdeg = ws;                          // [N]
    float* indeg  = outdeg + N;                  // [N]
    float* ns     = indeg + N;                   // [N]
    float* nd     = ns + N;                      // [N]
    float* h      = nd + N;                      // [N,64]
    float* agg    = h + (size_t)N * 64;          // [N,64]
    float* xb     = agg + (size_t)N * 64;        // [N,64]

    // degrees + norms
    gcn_zero4<<<cdiv(2LL * N / 4, 256), 256, 0, stream>>>((float4*)outdeg, 2 * N / 4);
    gcn_degrees<<<cdiv(E, 256), 256, 0, stream>>>(src, dst, outdeg, indeg, E);
    gcn_norms<<<cdiv(N, 256), 256, 0, stream>>>(outdeg, indeg, ns, nd, N);

    const int gBlocks = cdiv(N, 32);

    // ---- layer 0: 128 -> 64, relu ----
    gcn_gemm_wmma<128, 64><<<gBlocks, 32 * 2 * 4, 0, stream>>>(feat, ns, W0, h, N);
    gcn_zero4<<<cdiv((long long)N * 16, 256), 256, 0, stream>>>((float4*)agg, N * 16);
    gcn_scatter<64><<<cdiv((long long)E * 16, 256), 256, 0, stream>>>(h, src, dst, agg, E);
    gcn_finalize<64, true><<<cdiv((long long)N * 16, 256), 256, 0, stream>>>(agg, nd, b0, xb, N);

    // ---- layer 1: 64 -> 64, relu ----
    gcn_gemm_wmma<64, 64><<<gBlocks, 32 * 2 * 4, 0, stream>>>(xb, ns, W1, h, N);
    gcn_zero4<<<cdiv((long long)N * 16, 256), 256, 0, stream>>>((float4*)agg, N * 16);
    gcn_scatter<64><<<cdiv((long long)E * 16, 256), 256, 0, stream>>>(h, src, dst, agg, E);
    gcn_finalize<64, true><<<cdiv((long long)N * 16, 256), 256, 0, stream>>>(agg, nd, b1, xb, N);

    // ---- layer 2: 64 -> 64, relu ----
    gcn_gemm_wmma<64, 64><<<gBlocks, 32 * 2 * 4, 0, stream>>>(xb, ns, W2, h, N);
    gcn_zero4<<<cdiv((long long)N * 16, 256), 256, 0, stream>>>((float4*)agg, N * 16);
    gcn_scatter<64><<<cdiv((long long)E * 16, 256), 256, 0, stream>>>(h, src, dst, agg, E);
    gcn_finalize<64, true><<<cdiv((long long)N * 16, 256), 256, 0, stream>>>(agg, nd, b2, xb, N);

    // ---- layer 3: 64 -> 40, no relu, writes d_out ----
    gcn_gemm_wmma<64, 40><<<gBlocks, 32 * 2 * 3, 0, stream>>>(xb, ns, W3, h, N);
    gcn_zero4<<<cdiv((long long)N * 10, 256), 256, 0, stream>>>((float4*)agg, N * 10);
    gcn_scatter<40><<<cdiv((long long)E * 10, 256), 256, 0, stream>>>(h, src, dst, agg, E);
    gcn_finalize<40, false><<<cdiv((long long)N * 10, 256), 256, 0, stream>>>(agg, nd, b3, (float*)d_out, N);
}